// FlashAttention_50130858279245
// MI455X (gfx1250) — compile-verified
//
#include <hip/hip_runtime.h>
#include <math.h>

typedef __attribute__((ext_vector_type(2))) float v2f;
typedef __attribute__((ext_vector_type(8))) float v8f;

#define BATCH 2
#define SEQ 2048
#define HIDDEN 1024
#define HEADS 16
#define DHEAD 64
#define BHSD (BATCH * HEADS * SEQ * DHEAD)   // 4,194,304 floats per q/k/v

#define KV_PAD 76                            // 16B-aligned rows, conflict-free col reads
#define P_PAD 17

static __device__ __forceinline__ v8f wmma_f32(v2f a, v2f b, v8f c) {
    return __builtin_amdgcn_wmma_f32_16x16x4_f32(false, a, false, b, (short)0, c,
                                                 false, false);
}

// ---------------------------------------------------------------------------
// Kernel 1: QKV projection.  C[4096x3072] = X[4096x1024] @ W[1024x3072],
// scatter-stored into qkv workspace laid out as [3][B][H][S][64].
// One wave computes a 32x64 tile (2M x 4N WMMA sub-tiles): 8 WMMA per k-step
// fed by 2 b64 + 8 b32 loads.
// ---------------------------------------------------------------------------
__global__ void qkv_gemm_kernel(const float* __restrict__ X,
                                const float* __restrict__ W,
                                float* __restrict__ qkv) {
    const int N = 3 * HIDDEN;          // 3072
    const int K = HIDDEN;              // 1024

    int wave = (blockIdx.x * blockDim.x + threadIdx.x) >> 5;
    int lane = threadIdx.x & 31;
    int half = lane >> 4;
    int l16 = lane & 15;

    int ntn = N / 64;                  // 48
    int tm = wave / ntn;
    int tn = wave % ntn;
    int m0 = tm * 32;
    int n0 = tn * 64;

    const float* arow0 = X + (size_t)(m0 + l16) * K;
    const float* arow1 = arow0 + (size_t)16 * K;
    const float* bcol = W + n0 + l16;

    v8f acc[2][4] = {{{}, {}, {}, {}}, {{}, {}, {}, {}}};
    for (int k = 0; k < K; k += 4) {
        int kk = k + 2 * half;
        v2f a0 = *(const v2f*)(arow0 + kk);
        v2f a1 = *(const v2f*)(arow1 + kk);
        const float* b0 = bcol + (size_t)kk * N;
        const float* b1 = bcol + (size_t)(kk + 1) * N;
#pragma unroll
        for (int nt = 0; nt < 4; ++nt) {
            v2f b;
            b.x = b0[nt * 16];
            b.y = b1[nt * 16];
            acc[0][nt] = wmma_f32(a0, b, acc[0][nt]);
            acc[1][nt] = wmma_f32(a1, b, acc[1][nt]);
        }
    }

#pragma unroll
    for (int nt = 0; nt < 4; ++nt) {
        int col = n0 + nt * 16 + l16;
        int which = col >> 10;
        int rem = col & 1023;
        int head = rem >> 6;
        int d = rem & 63;
#pragma unroll
        for (int mt = 0; mt < 2; ++mt)
#pragma unroll
            for (int i = 0; i < 8; ++i) {
                int m = m0 + mt * 16 + i + 8 * half;
                int bb = m >> 11;                        // / SEQ
                int s = m & 2047;                        // % SEQ
                size_t idx =
                    ((((size_t)which * BATCH + bb) * HEADS + head) * SEQ + s) * DHEAD + d;
                qkv[idx] = acc[mt][nt][i];
            }
    }
}

// ---------------------------------------------------------------------------
// Kernel 2: flash attention.  One block owns one (b,h) x 128 query rows:
// 8 waves x 16 rows.  K/V tiles staged cooperatively in LDS per 16-key step.
// q/k/v are [B*H][S][64].  ctx output is [B][S][HIDDEN].
// ---------------------------------------------------------------------------
__global__ void attn_kernel(const float* __restrict__ q,
                            const float* __restrict__ k,
                            const float* __restrict__ v,
                            float* __restrict__ ctx) {
    __shared__ float ldsK[16 * KV_PAD];
    __shared__ float ldsV[16 * KV_PAD];
    __shared__ float lds_p[8][16 * P_PAD];

    int tid = threadIdx.x;
    int wv = tid >> 5;
    int lane = tid & 31;
    int half = lane >> 4;
    int l16 = lane & 15;

    int bh = blockIdx.x >> 4;                            // / (SEQ/128)
    int qchunk = blockIdx.x & 15;
    int qb = qchunk * 8 + wv;                            // 16-row query block

    // cooperative staging coords: 256 threads x float4 = 16x64 tile
    int srow = tid >> 4;                                 // 0..15
    int scol = (tid & 15) * 4;                           // 0,4,...,60

    const float scale = 0.125f;                          // 1/sqrt(64)

    const float* qrow = q + ((size_t)bh * SEQ + qb * 16 + l16) * DHEAD;
    v2f qa[16];
#pragma unroll
    for (int t = 0; t < 16; ++t)
        qa[t] = *(const v2f*)(qrow + 4 * t + 2 * half);

    v8f acc[4] = {{}, {}, {}, {}};
    float m_row[8], l_row[8];
#pragma unroll
    for (int i = 0; i < 8; ++i) { m_row[i] = -INFINITY; l_row[i] = 0.0f; }

    const float* kstage = k + (size_t)bh * SEQ * DHEAD + (size_t)srow * DHEAD + scol;
    const float* vstage = v + (size_t)bh * SEQ * DHEAD + (size_t)srow * DHEAD + scol;

    for (int j = 0; j < SEQ / 16; ++j) {
        // ---- stage K_j, V_j (16x64 each) into LDS ----
        float4 kv4 = *(const float4*)(kstage + (size_t)j * 16 * DHEAD);
        float4 vv4 = *(const float4*)(vstage + (size_t)j * 16 * DHEAD);
        *(float4*)&ldsK[srow * KV_PAD + scol] = kv4;
        *(float4*)&ldsV[srow * KV_PAD + scol] = vv4;
        __syncthreads();

        // ---- S = Q @ K_j^T  (16 WMMA k-steps over D=64) ----
        const float* krow = &ldsK[l16 * KV_PAD];         // B: N = l16
        v8f c = {};
#pragma unroll
        for (int t = 0; t < 16; ++t) {
            v2f b = *(const v2f*)(krow + 4 * t + 2 * half);
            c = wmma_f32(qa[t], b, c);
        }

        // ---- online softmax update (rows live in 16-lane halves) ----
        float alpha[8];
#pragma unroll
        for (int i = 0; i < 8; ++i) {
            float sc = c[i] * scale;
            float mx = sc;
#pragma unroll
            for (int off = 8; off > 0; off >>= 1)
                mx = fmaxf(mx, __shfl_xor(mx, off, 16));
            float m_new = fmaxf(m_row[i], mx);
            alpha[i] = __expf(m_row[i] - m_new);
            float p = __expf(sc - m_new);
            float ps = p;
#pragma unroll
            for (int off = 8; off > 0; off >>= 1)
                ps += __shfl_xor(ps, off, 16);
            l_row[i] = l_row[i] * alpha[i] + ps;
            m_row[i] = m_new;
            lds_p[wv][(i + 8 * half) * P_PAD + l16] = p;
        }
#pragma unroll
        for (int dt = 0; dt < 4; ++dt)
#pragma unroll
            for (int i = 0; i < 8; ++i) acc[dt][i] *= alpha[i];

        asm volatile("s_wait_dscnt 0" ::: "memory");
        __builtin_amdgcn_wave_barrier();

        // ---- acc += P @ V_j  (K=16 -> 4 k-steps; D=64 -> 4 output tiles) ----
#pragma unroll
        for (int t = 0; t < 4; ++t) {
            int kk = 4 * t + 2 * half;
            v2f a;                                       // A: M = l16
            a.x = lds_p[wv][l16 * P_PAD + kk];
            a.y = lds_p[wv][l16 * P_PAD + kk + 1];
            const float* vr = &ldsV[kk * KV_PAD + l16];
#pragma unroll
            for (int dt = 0; dt < 4; ++dt) {
                v2f b;                                   // B: N = l16 within tile dt
                b.x = vr[dt * 16];
                b.y = vr[dt * 16 + KV_PAD];
                acc[dt] = wmma_f32(a, b, acc[dt]);
            }
        }

        asm volatile("s_wait_dscnt 0" ::: "memory");
        __builtin_amdgcn_wave_barrier();
        __syncthreads();                                 // protect K/V restage
    }

    // ---- normalize and store ctx[B][S][HIDDEN] ----
    int bb = bh >> 4;
    int h = bh & 15;
#pragma unroll
    for (int i = 0; i < 8; ++i) {
        float inv_l = 1.0f / l_row[i];
        int s = qb * 16 + i + 8 * half;
#pragma unroll
        for (int dt = 0; dt < 4; ++dt) {
            int col = h * DHEAD + dt * 16 + l16;
            ctx[((size_t)bb * SEQ + s) * HIDDEN + col] = acc[dt][i] * inv_l;
        }
    }
}

// ---------------------------------------------------------------------------
// Kernel 3: output projection.  out[4096x1024] = ctx[4096x1024] @ W_o[1024x1024]
// One wave computes a 32x64 tile (2M x 4N sub-tiles).
// ---------------------------------------------------------------------------
__global__ void o_gemm_kernel(const float* __restrict__ A,
                              const float* __restrict__ W,
                              float* __restrict__ out) {
    const int N = HIDDEN;
    const int K = HIDDEN;

    int wave = (blockIdx.x * blockDim.x + threadIdx.x) >> 5;
    int lane = threadIdx.x & 31;
    int half = lane >> 4;
    int l16 = lane & 15;

    int ntn = N / 64;                  // 16
    int tm = wave / ntn;
    int tn = wave % ntn;
    int m0 = tm * 32;
    int n0 = tn * 64;

    const float* arow0 = A + (size_t)(m0 + l16) * K;
    const float* arow1 = arow0 + (size_t)16 * K;
    const float* bcol = W + n0 + l16;

    v8f acc[2][4] = {{{}, {}, {}, {}}, {{}, {}, {}, {}}};
    for (int k = 0; k < K; k += 4) {
        int kk = k + 2 * half;
        v2f a0 = *(const v2f*)(arow0 + kk);
        v2f a1 = *(const v2f*)(arow1 + kk);
        const float* b0 = bcol + (size_t)kk * N;
        const float* b1 = bcol + (size_t)(kk + 1) * N;
#pragma unroll
        for (int nt = 0; nt < 4; ++nt) {
            v2f b;
            b.x = b0[nt * 16];
            b.y = b1[nt * 16];
            acc[0][nt] = wmma_f32(a0, b, acc[0][nt]);
            acc[1][nt] = wmma_f32(a1, b, acc[1][nt]);
        }
    }

#pragma unroll
    for (int nt = 0; nt < 4; ++nt) {
        int col = n0 + nt * 16 + l16;
#pragma unroll
        for (int mt = 0; mt < 2; ++mt)
#pragma unroll
            for (int i = 0; i < 8; ++i) {
                int m = m0 + mt * 16 + i + 8 * half;
                out[(size_t)m * N + col] = acc[mt][nt][i];
            }
    }
}

// ---------------------------------------------------------------------------
extern "C" void kernel_launch(void* const* d_in, const int* in_sizes, int n_in,
                              void* d_out, int out_size, void* d_ws, size_t ws_size,
                              hipStream_t stream) {
    const float* hidden = (const float*)d_in[0];   // [B,S,H]
    const float* w_qkv  = (const float*)d_in[1];   // [H, 3H]
    const float* w_o    = (const float*)d_in[2];   // [H, H]
    float* out = (float*)d_out;

    float* qkv = (float*)d_ws;                     // [3][B][H][S][64]
    float* qp = qkv;
    float* kp = qkv + (size_t)BHSD;
    float* vp = qkv + (size_t)2 * BHSD;
    float* ctx = qkv + (size_t)3 * BHSD;           // [B][S][HIDDEN]

    // Kernel 1: (4096/32)*(3072/64) = 6144 waves -> 768 blocks of 8 waves
    qkv_gemm_kernel<<<768, 256, 0, stream>>>(hidden, w_qkv, qkv);

    // Kernel 2: one block per (b,h, 128 query rows): 32*16 = 512 blocks
    attn_kernel<<<512, 256, 0, stream>>>(qp, kp, vp, ctx);

    // Kernel 3: (4096/32)*(1024/64) = 2048 waves -> 256 blocks
    o_gemm_kernel<<<256, 256, 0, stream>>>(ctx, w_o, out);
}